// QuantumProjection_24670292148264
// MI455X (gfx1250) — compile-verified
//
#include <hip/hip_runtime.h>

typedef __attribute__((ext_vector_type(2))) float v2f;
typedef __attribute__((ext_vector_type(8))) float v8f;

#define D_IN   1024
#define D_OUT  1024
#define NQ     4
#define THREADS 256           // 8 waves (wave32); block owns 256 rows

// CNOT ring permutation piece: psi' [j] = psi[ j ^ (bit(3-c) << (3-t)) ]
constexpr int cperm(int c, int t, int j) {
    return j ^ (((j >> (3 - c)) & 1) << (3 - t));
}

__global__ __launch_bounds__(THREADS)
void qproj_fused(const float* __restrict__ x,
                 const float* __restrict__ pre_w,
                 const float* __restrict__ pre_b,
                 const float* __restrict__ qw,
                 const float* __restrict__ post_w,
                 const float* __restrict__ post_b,
                 float* __restrict__ out)
{
    __shared__ float s_prew[NQ * D_IN];    // 16 KB  [q][k]
    __shared__ float s_postw[D_OUT * NQ];  // 16 KB  [j][q] (row-major, matches input)
    __shared__ float s_postb[D_OUT];       //  4 KB
    __shared__ float s_z[THREADS * NQ];    //  4 KB  per-row expectation values

    const int tid = threadIdx.x;

    for (int i = tid; i < NQ * D_IN;  i += THREADS) s_prew[i]  = pre_w[i];
    for (int i = tid; i < D_OUT * NQ; i += THREADS) s_postw[i] = post_w[i];
    for (int i = tid; i < D_OUT;      i += THREADS) s_postb[i] = post_b[i];
    __syncthreads();

    // ---------- Stage A: h = tanh(x @ pre_w.T + pre_b) * pi (one row per thread) ----------
    const long row = (long)blockIdx.x * THREADS + tid;
    const float* xr = x + row * D_IN;
    float acc0 = 0.f, acc1 = 0.f, acc2 = 0.f, acc3 = 0.f;
    #pragma unroll 8
    for (int k = 0; k < D_IN; k += 4) {
        float4 xv = *(const float4*)(xr + k);
        float4 w0 = *(const float4*)(s_prew + 0 * D_IN + k);
        float4 w1 = *(const float4*)(s_prew + 1 * D_IN + k);
        float4 w2 = *(const float4*)(s_prew + 2 * D_IN + k);
        float4 w3 = *(const float4*)(s_prew + 3 * D_IN + k);
        acc0 = fmaf(xv.x, w0.x, fmaf(xv.y, w0.y, fmaf(xv.z, w0.z, fmaf(xv.w, w0.w, acc0))));
        acc1 = fmaf(xv.x, w1.x, fmaf(xv.y, w1.y, fmaf(xv.z, w1.z, fmaf(xv.w, w1.w, acc1))));
        acc2 = fmaf(xv.x, w2.x, fmaf(xv.y, w2.y, fmaf(xv.z, w2.z, fmaf(xv.w, w2.w, acc2))));
        acc3 = fmaf(xv.x, w3.x, fmaf(xv.y, w3.y, fmaf(xv.z, w3.z, fmaf(xv.w, w3.w, acc3))));
    }

    // ---------- Stage B: 4-qubit circuit, fully in registers ----------
    const float PI = 3.14159265358979323846f;
    const float RSQRT2 = 0.70710678118654752f;
    float hacc[NQ] = { acc0 + pre_b[0], acc1 + pre_b[1], acc2 + pre_b[2], acc3 + pre_b[3] };

    // Per-qubit prep: v = RZ(atan(f^2)) * RY(atan(f)) * H * |0>  (product state)
    float vr[NQ][2], vi[NQ][2];
    #pragma unroll
    for (int q = 0; q < NQ; ++q) {
        float f = tanhf(hacc[q]) * PI;
        float a = atanf(f);
        float b = atanf(f * f);
        float sa, ca; sincosf(0.5f * a, &sa, &ca);
        float sb, cb; sincosf(0.5f * b, &sb, &cb);
        float t0 = (ca - sa) * RSQRT2;
        float t1 = (ca + sa) * RSQRT2;
        vr[q][0] =  t0 * cb;  vi[q][0] = -t0 * sb;
        vr[q][1] =  t1 * cb;  vi[q][1] =  t1 * sb;
    }

    // Kron: psi[j] = v0[b3] v1[b2] v2[b1] v3[b0]  (qubit 0 = MSB of j)
    float r2[4], i2[4];
    #pragma unroll
    for (int a0 = 0; a0 < 2; ++a0)
        #pragma unroll
        for (int a1 = 0; a1 < 2; ++a1) {
            r2[a0 * 2 + a1] = vr[0][a0] * vr[1][a1] - vi[0][a0] * vi[1][a1];
            i2[a0 * 2 + a1] = vr[0][a0] * vi[1][a1] + vi[0][a0] * vr[1][a1];
        }
    float r3[8], i3[8];
    #pragma unroll
    for (int p = 0; p < 4; ++p)
        #pragma unroll
        for (int a2 = 0; a2 < 2; ++a2) {
            r3[p * 2 + a2] = r2[p] * vr[2][a2] - i2[p] * vi[2][a2];
            i3[p * 2 + a2] = r2[p] * vi[2][a2] + i2[p] * vr[2][a2];
        }
    float pr[16], pi_[16];
    #pragma unroll
    for (int p = 0; p < 8; ++p)
        #pragma unroll
        for (int a3 = 0; a3 < 2; ++a3) {
            pr[p * 2 + a3] = r3[p] * vr[3][a3] - i3[p] * vi[3][a3];
            pi_[p * 2 + a3] = r3[p] * vi[3][a3] + i3[p] * vr[3][a3];
        }

    // CNOT ring (4 perms composed at compile time into one register shuffle)
    float qr[16], qi[16];
    #pragma unroll
    for (int j = 0; j < 16; ++j) {
        const int s = cperm(0, 1, cperm(1, 2, cperm(2, 3, cperm(3, 0, j))));
        qr[j] = pr[s];  qi[j] = pi_[s];
    }

    // Rot(phi, th, om) on each qubit; weights are batch-uniform
    #pragma unroll
    for (int q = 0; q < NQ; ++q) {
        float phi = qw[q * 3 + 0], th = qw[q * 3 + 1], om = qw[q * 3 + 2];
        float s, c;   sincosf(0.5f * th, &s, &c);
        float sp, cp; sincosf(0.5f * (phi + om), &sp, &cp);
        float sq, cq; sincosf(0.5f * (phi - om), &sq, &cq);
        float u00r =  c * cp, u00i = -c * sp;
        float u01r = -s * cq, u01i = -s * sq;
        float u10r =  s * cq, u10i = -s * sq;
        float u11r =  c * cp, u11i =  c * sp;
        const int m = 1 << (3 - q);
        #pragma unroll
        for (int j = 0; j < 16; ++j) {
            if (j & m) continue;
            const int j1 = j | m;
            float ar = qr[j],  ai = qi[j];
            float br = qr[j1], bi = qi[j1];
            qr[j]  = u00r * ar - u00i * ai + u01r * br - u01i * bi;
            qi[j]  = u00r * ai + u00i * ar + u01r * bi + u01i * br;
            qr[j1] = u10r * ar - u10i * ai + u11r * br - u11i * bi;
            qi[j1] = u10r * ai + u10i * ar + u11r * bi + u11i * br;
        }
    }

    // z_i = sum_j (1 - 2*bit_{3-i}(j)) |psi_j|^2
    float z0 = 0.f, z1 = 0.f, z2 = 0.f, z3 = 0.f;
    #pragma unroll
    for (int j = 0; j < 16; ++j) {
        float p = qr[j] * qr[j] + qi[j] * qi[j];
        z0 += ((j >> 3) & 1) ? -p : p;
        z1 += ((j >> 2) & 1) ? -p : p;
        z2 += ((j >> 1) & 1) ? -p : p;
        z3 += ((j >> 0) & 1) ? -p : p;
    }
    s_z[tid * 4 + 0] = z0;
    s_z[tid * 4 + 1] = z1;
    s_z[tid * 4 + 2] = z2;
    s_z[tid * 4 + 3] = z3;
    __syncthreads();

    // ---------- Stage C: out = z @ post_w.T + post_b via V_WMMA_F32_16X16X4_F32 ----------
    // A (16x4 f32):  v0 = K0 (lanes 0-15) / K2 (lanes 16-31); v1 = K1 / K3; M = lane&15
    // B (4x16 f32):  v0 = K0 row (lanes 0-15) / K2 row; v1 = K1 / K3; N = lane&15
    // D (16x16 f32): vr = row r (lanes 0-15) / row r+8 (lanes 16-31); N = lane&15
    const int wave = tid >> 5;
    const int lane = tid & 31;
    const int lh   = lane >> 4;   // 0 or 1
    const int lm   = lane & 15;

    #pragma unroll
    for (int t = 0; t < 2; ++t) {
        const int rbase = (wave * 2 + t) * 16;     // row tile within block
        const float* zp = &s_z[(rbase + lm) * 4 + lh * 2];
        v2f A;  A.x = zp[0];  A.y = zp[1];
        const long growr = (long)blockIdx.x * THREADS + rbase;

        for (int cb = 0; cb < D_OUT / 16; ++cb) {
            const int col = cb * 16 + lm;
            float4 pw = *(const float4*)(s_postw + col * 4);   // post_w[col][0..3]
            v2f Bv;  Bv.x = lh ? pw.z : pw.x;  Bv.y = lh ? pw.w : pw.y;
            float pb = s_postb[col];
            v8f C = { pb, pb, pb, pb, pb, pb, pb, pb };

            v8f D = __builtin_amdgcn_wmma_f32_16x16x4_f32(
                        false, A, false, Bv, (short)0, C, false, false);

            float* op = out + (growr + lh * 8) * D_OUT + col;
            #pragma unroll
            for (int r = 0; r < 8; ++r)
                __builtin_nontemporal_store(D[r], op + (long)r * D_OUT);
        }
    }
}

extern "C" void kernel_launch(void* const* d_in, const int* in_sizes, int n_in,
                              void* d_out, int out_size, void* d_ws, size_t ws_size,
                              hipStream_t stream) {
    const float* x      = (const float*)d_in[0];
    const float* pre_w  = (const float*)d_in[1];
    const float* pre_b  = (const float*)d_in[2];
    const float* qw     = (const float*)d_in[3];
    const float* post_w = (const float*)d_in[4];
    const float* post_b = (const float*)d_in[5];
    float* out = (float*)d_out;

    const int nrows  = in_sizes[0] / D_IN;     // 65536
    const int blocks = nrows / THREADS;        // 256
    qproj_fused<<<blocks, THREADS, 0, stream>>>(x, pre_w, pre_b, qw, post_w, post_b, out);
}